// DefCorFixW_71786083385911
// MI455X (gfx1250) — compile-verified
//
#include <hip/hip_runtime.h>

typedef __attribute__((ext_vector_type(2))) float v2f;
typedef __attribute__((ext_vector_type(4))) float v4f;
typedef __attribute__((ext_vector_type(8))) float v8f;

#define B_  8
#define C_  128
#define H_  96
#define W_  96
#define HW_ (H_ * W_)
#define T_  9
#define K_  9

// One block: fixed (b, kpos), 128 spatial positions (8 waves x 16).
// Each wave: 16x16x128 GEMM tile via 32x v_wmma_f32_16x16x4_f32,
// A = bilinear-gathered samples (f32), B = weight[:, :, kpos] from LDS.
__global__ __launch_bounds__(256) void defcor_wmma_kernel(
    const float* __restrict__ x,       // (B, C, H, W)
    const float* __restrict__ offset,  // (B, 2K, H, W)
    const float* __restrict__ weight,  // (C, T, 3, 3)
    float* __restrict__ out)           // (B, T*K, H, W)
{
    __shared__ float wlds[C_ * 16];   // weight[:, :, kpos], t padded 9->16 with zeros

    const int bid  = blockIdx.x;
    const int bk   = bid / 72;        // 72 spatial blocks of 128 per (b,k)
    const int sblk = bid % 72;
    const int b    = bk / K_;
    const int kpos = bk % K_;

    const int tid = threadIdx.x;

    // Stage B-panel into LDS: wlds[c*16 + n] = weight[c, n, kpos] (n<9), else 0.
    for (int i = tid; i < C_ * 16; i += 256) {
        const int c = i >> 4;
        const int n = i & 15;
        wlds[i] = (n < T_) ? weight[(size_t)(c * T_ + n) * K_ + kpos] : 0.0f;
    }
    __syncthreads();

    const int wave = tid >> 5;
    const int lane = tid & 31;
    const int half = lane >> 4;       // selects K-pair {0,1} vs {2,3} of the 16x4 A tile
    const int mloc = lane & 15;       // M index within tile

    const int sp0 = sblk * 128 + wave * 16;   // first spatial index of this wave's tile
    const int s   = sp0 + mloc;
    const int h   = s / W_;
    const int w   = s % W_;

    // Deformed sample position (STRIDE=1, PAD=1, DIL=1)
    const float offy = offset[((size_t)((b * K_ + kpos) * 2 + 0)) * HW_ + s];
    const float offx = offset[((size_t)((b * K_ + kpos) * 2 + 1)) * HW_ + s];
    const float py = (float)(h - 1 + kpos / 3) + offy;
    const float px = (float)(w - 1 + kpos % 3) + offx;

    const float y0f = floorf(py), x0f = floorf(px);
    const float ly = py - y0f,    lx = px - x0f;
    const int   y0 = (int)y0f,    x0 = (int)x0f;
    const int   y1 = y0 + 1,      x1 = x0 + 1;

    const float vy0 = (y0 >= 0 && y0 < H_) ? 1.0f : 0.0f;
    const float vy1 = (y1 >= 0 && y1 < H_) ? 1.0f : 0.0f;
    const float vx0 = (x0 >= 0 && x0 < W_) ? 1.0f : 0.0f;
    const float vx1 = (x1 >= 0 && x1 < W_) ? 1.0f : 0.0f;

    // corner bilinear weights with validity folded in (reference: g * valid)
    const float wv00 = (1.0f - ly) * (1.0f - lx) * vy0 * vx0;
    const float wv01 = (1.0f - ly) * lx          * vy0 * vx1;
    const float wv10 = ly          * (1.0f - lx) * vy1 * vx0;
    const float wv11 = ly          * lx          * vy1 * vx1;

    const int cy0 = y0 < 0 ? 0 : (y0 >= H_ ? H_ - 1 : y0);
    const int cy1 = y1 < 0 ? 0 : (y1 >= H_ ? H_ - 1 : y1);
    const int cx0 = x0 < 0 ? 0 : (x0 >= W_ ? W_ - 1 : x0);
    const int cx1 = x1 < 0 ? 0 : (x1 >= W_ ? W_ - 1 : x1);

    // Corner base pointers include this lane's channel-pair offset (2*half planes).
    const float* __restrict__ xb  = x + ((size_t)b * C_ + 2 * half) * HW_;
    const float* __restrict__ p00 = xb + cy0 * W_ + cx0;
    const float* __restrict__ p01 = xb + cy0 * W_ + cx1;
    const float* __restrict__ p10 = xb + cy1 * W_ + cx0;
    const float* __restrict__ p11 = xb + cy1 * W_ + cx1;

    // B-panel base for this lane: wlds[(c0 + 2*half)*16 + n]
    const float* __restrict__ wl = &wlds[(2 * half) * 16 + mloc];

    v8f acc = {};
#pragma unroll
    for (int c0 = 0; c0 < C_; c0 += 4) {
        const size_t co = (size_t)c0 * HW_;   // constant per unrolled iter -> imm offset
        const float a0 = wv00 * p00[co]       + wv01 * p01[co]
                       + wv10 * p10[co]       + wv11 * p11[co];
        const float a1 = wv00 * p00[co + HW_] + wv01 * p01[co + HW_]
                       + wv10 * p10[co + HW_] + wv11 * p11[co + HW_];
        v2f A = {a0, a1};
        v2f Bm = {wl[c0 * 16], wl[c0 * 16 + 16]};
        // D = A(16x4) x B(4x16) + C, all f32
        acc = __builtin_amdgcn_wmma_f32_16x16x4_f32(
            false, A, false, Bm, (short)0, acc, false, false);
    }

    // D layout: VGPR r -> M = r + 8*half, N = lane&15. N is the t index.
    if (mloc < T_) {
        const int tchan = mloc * K_ + kpos;                // out channel = t*K + k
        float* __restrict__ ob =
            out + ((size_t)b * (T_ * K_) + tchan) * HW_ + sp0 + 8 * half;
        v4f lo = {acc[0], acc[1], acc[2], acc[3]};
        v4f hi = {acc[4], acc[5], acc[6], acc[7]};
        *(v4f*)(ob)     = lo;
        *(v4f*)(ob + 4) = hi;
    }
}

extern "C" void kernel_launch(void* const* d_in, const int* in_sizes, int n_in,
                              void* d_out, int out_size, void* d_ws, size_t ws_size,
                              hipStream_t stream) {
    const float* x      = (const float*)d_in[0];
    const float* offset = (const float*)d_in[1];
    const float* weight = (const float*)d_in[2];
    float* out          = (float*)d_out;

    // 8 batches * 9 kernel positions * 72 spatial blocks (128 positions each)
    const int grid = B_ * K_ * 72;   // 5184
    defcor_wmma_kernel<<<grid, 256, 0, stream>>>(x, offset, weight, out);
}